// MatchAttention_4655744549272
// MI455X (gfx1250) — compile-verified
//
#include <hip/hip_runtime.h>
#include <math.h>

typedef __attribute__((ext_vector_type(2))) float v2f;
typedef __attribute__((ext_vector_type(8))) float v8f;

// ---------------------------------------------------------------------------
// One wave computes a 16x16 f32 tile of C = A(16xK) * B(Kx16) with
// V_WMMA_F32_16X16X4_F32, stepping K by 4. Row-major A (lda) / B (ldb).
// Fragment layouts per CDNA5 ISA 7.12.2 (32-bit A 16x4):
//   lanes 0-15 : M=lane,   VGPR0=K+0, VGPR1=K+1
//   lanes16-31 : M=lane-16,VGPR0=K+2, VGPR1=K+3
// B mirrored with N=lane&15.
// ---------------------------------------------------------------------------
__device__ inline v8f wmma_f32_tile(const float* __restrict__ A, int lda,
                                    const float* __restrict__ B, int ldb,
                                    int K) {
  const int t   = threadIdx.x & 31;
  const int mn  = t & 15;          // A: row M, B: col N
  const int klo = (t >> 4) << 1;   // 0 for lanes 0-15, 2 for lanes 16-31
  v8f acc = {};
  for (int k = 0; k < K; k += 4) {
    v2f a = *(const v2f*)(A + mn * lda + k + klo);   // contiguous K pair
    v2f b;
    const float* bp = B + (k + klo) * ldb + mn;
    b.x = bp[0];
    b.y = bp[ldb];
    // 8 args: (neg_a, A, neg_b, B, c_mod, C, reuse_a, reuse_b)
    acc = __builtin_amdgcn_wmma_f32_16x16x4_f32(false, a, false, b,
                                                (short)0, acc, false, false);
  }
  return acc;
}

// C/D layout: VGPR i -> row (i + 8*(lane>=16)), col (lane&15)
__device__ inline void wmma_store_tile(float* __restrict__ C, int ldc,
                                       v8f acc, float addend) {
  const int t    = threadIdx.x & 31;
  const int col  = t & 15;
  const int row0 = (t >> 4) << 3;
#pragma unroll
  for (int i = 0; i < 8; ++i)
    C[(row0 + i) * ldc + col] = acc[i] + addend;
}

// ---------------------------------------------------------------------------
// Kernel A: PL = left @ Wl + b   (rows 0..1023)
//           PR = right @ Wr      (rows 1024..2047)
// 2048x256 output over K=256; 1 wave per 16x16 tile -> 128x16 = 2048 waves.
// ---------------------------------------------------------------------------
__global__ void __launch_bounds__(256)
proj_gemm_kernel(const float* __restrict__ left,
                 const float* __restrict__ right,
                 const float* __restrict__ W,
                 const float* __restrict__ bias,
                 float* __restrict__ PL, float* __restrict__ PR) {
  const int wave = (int)((blockIdx.x * blockDim.x + threadIdx.x) >> 5);
  const int tn = wave & 15;   // 16 col tiles over F=256
  const int tm = wave >> 4;   // 128 row tiles over 2048 rows
  const float* Asrc;
  const float* Bsrc;
  float* Cdst;
  float add;
  if (tm < 64) {
    Asrc = left + tm * 16 * 256;
    Bsrc = W;                         // Wl = W[0:256,:]
    Cdst = PL + tm * 16 * 256;
    add  = bias[tn * 16 + (threadIdx.x & 15)];  // fold bias into PL once
  } else {
    Asrc = right + (tm - 64) * 16 * 256;
    Bsrc = W + 256 * 256;             // Wr = W[256:512,:]
    Cdst = PR + (tm - 64) * 16 * 256;
    add  = 0.0f;
  }
  v8f acc = wmma_f32_tile(Asrc, 256, Bsrc + tn * 16, 256, 256);
  wmma_store_tile(Cdst + tn * 16, 256, acc, add);
}

// ---------------------------------------------------------------------------
// tanh on the gfx1250 hardware transcendental path (V_TANH_F32) when the
// builtin exists; otherwise stable exp-based fallback (v_exp_f32 + rcp).
// ---------------------------------------------------------------------------
__device__ inline float fast_tanh(float x) {
#if __has_builtin(__builtin_amdgcn_tanhf)
  return __builtin_amdgcn_tanhf(x);
#else
  float ax = fabsf(x);
  float e  = __expf(-2.0f * ax);
  float t  = (1.0f - e) / (1.0f + e);
  return copysignf(t, x);
#endif
}

// ---------------------------------------------------------------------------
// Kernel B (fused, never materializes the (B,L,R,F) hid tensor):
//   one block per (b,l); thread r in [0,256):
//   score[r] = sum_f tanh(PL[b,l,f] + PR[b,r,f]) * v[f] + b2
//   weights[b,l,:] = softmax(score)
// PL row + v staged in LDS (128-bit ds loads); PR streamed as float4
// (global_load_b128), hot in L2 (256KB/batch reused by 256 blocks).
// ---------------------------------------------------------------------------
__global__ void __launch_bounds__(256)
score_softmax_kernel(const float* __restrict__ PL,
                     const float* __restrict__ PR,
                     const float* __restrict__ v,
                     const float* __restrict__ b2,
                     float* __restrict__ wt) {
  const int bl = (int)blockIdx.x;        // b*256 + l
  const int b  = bl >> 8;
  const int r  = (int)threadIdx.x;

  __shared__ __align__(16) float s_pl[256];
  __shared__ __align__(16) float s_v[256];
  __shared__ float red[256];

  s_pl[r] = PL[bl * 256 + r];
  s_v[r]  = v[r];
  __syncthreads();

  const float4* __restrict__ pr4 = (const float4*)(PR + (b * 256 + r) * 256);
  const float4* s_pl4 = (const float4*)s_pl;
  const float4* s_v4  = (const float4*)s_v;

  float s = 0.0f;
#pragma unroll 4
  for (int f = 0; f < 64; ++f) {
    float4 p  = pr4[f];       // global_load_b128, L2-resident
    float4 q  = s_pl4[f];     // ds_load_b128
    float4 vv = s_v4[f];      // ds_load_b128
    s = fmaf(fast_tanh(q.x + p.x), vv.x, s);
    s = fmaf(fast_tanh(q.y + p.y), vv.y, s);
    s = fmaf(fast_tanh(q.z + p.z), vv.z, s);
    s = fmaf(fast_tanh(q.w + p.w), vv.w, s);
  }
  s += b2[0];

  // block softmax over the 256 r's
  red[r] = s;
  __syncthreads();
  for (int off = 128; off > 0; off >>= 1) {
    if (r < off) red[r] = fmaxf(red[r], red[r + off]);
    __syncthreads();
  }
  const float mx = red[0];
  __syncthreads();
  const float e = __expf(s - mx);
  red[r] = e;
  __syncthreads();
  for (int off = 128; off > 0; off >>= 1) {
    if (r < off) red[r] += red[r + off];
    __syncthreads();
  }
  wt[bl * 256 + r] = e / red[0];
}

// ---------------------------------------------------------------------------
// Kernel C: out[b] = weights[b] (256x256) @ right[b] (256x256)
// 4 batches * 16x16 tiles = 1024 waves.
// ---------------------------------------------------------------------------
__global__ void __launch_bounds__(256)
attend_gemm_kernel(const float* __restrict__ wt,
                   const float* __restrict__ right,
                   float* __restrict__ out) {
  const int wave  = (int)((blockIdx.x * blockDim.x + threadIdx.x) >> 5);
  const int tn    = wave & 15;        // col tiles over D=256
  const int tmb   = wave >> 4;        // batch*16 + row tile
  const int batch = tmb >> 4;
  const int tm    = tmb & 15;
  const float* A = wt    + batch * 256 * 256 + tm * 16 * 256;
  const float* B = right + batch * 256 * 256 + tn * 16;
  float*       C = out   + batch * 256 * 256 + tm * 16 * 256 + tn * 16;
  v8f acc = wmma_f32_tile(A, 256, B, 256, 256);
  wmma_store_tile(C, 256, acc, 0.0f);
}

extern "C" void kernel_launch(void* const* d_in, const int* in_sizes, int n_in,
                              void* d_out, int out_size, void* d_ws,
                              size_t ws_size, hipStream_t stream) {
  (void)in_sizes; (void)n_in; (void)out_size; (void)ws_size;
  const float* left  = (const float*)d_in[0];  // (4,256,256)
  const float* right = (const float*)d_in[1];  // (4,256,256)
  const float* W     = (const float*)d_in[2];  // (512,256)
  const float* bias  = (const float*)d_in[3];  // (256,)
  const float* v     = (const float*)d_in[4];  // (256,)
  const float* b2    = (const float*)d_in[5];  // (1,)
  float* out = (float*)d_out;                  // (4,256,256)

  float* ws = (float*)d_ws;
  float* PL = ws;                 // 4*256*256 floats
  float* PR = ws + 262144;        // 4*256*256 floats
  float* WT = ws + 524288;        // 4*256*256 floats

  // A: projections via f32 WMMA — 2048 waves = 256 blocks * 8 waves
  proj_gemm_kernel<<<256, 256, 0, stream>>>(left, right, W, bias, PL, PR);

  // B: fused tanh-score + softmax — one block per (b,l)
  score_softmax_kernel<<<1024, 256, 0, stream>>>(PL, PR, v, b2, WT);

  // C: weights @ right via f32 WMMA — 1024 waves = 128 blocks * 8 waves
  attend_gemm_kernel<<<128, 256, 0, stream>>>(WT, right, out);
}